// GSNET_semseg_s3dis_61890478735452
// MI455X (gfx1250) — compile-verified
//
#include <hip/hip_runtime.h>
#include <hip/hip_bf16.h>

#define KNN 20

typedef __attribute__((ext_vector_type(2))) float v2f;
typedef __attribute__((ext_vector_type(8))) float v8f;

// ---------------------------------------------------------------------------
// Layout convention: features stored as [C][B*N] row-major, element (b,c,n)
// at  c*(B*N) + b*N + n.  Grouped tensors: [C][B*N*KNN], col = (b*N+n)*KNN+j.
// ---------------------------------------------------------------------------

__global__ void repack_pts1_kernel(const float* __restrict__ x, float* __restrict__ p1,
                                   int B, int N) {
  int g = blockIdx.x * blockDim.x + threadIdx.x;
  int BN = B * N;
  if (g >= 3 * BN) return;
  int c = g / BN, r = g - c * BN;
  int b = r / N, n = r - b * N;
  p1[g] = x[((size_t)b * 9 + c) * N + n];   // x is (B, 9, N)
}

__global__ void pack_weight_kernel(const float* __restrict__ w, int Cout, int Cin,
                                   int Coutp, int Cinp, float* __restrict__ dst) {
  int g = blockIdx.x * blockDim.x + threadIdx.x;
  if (g >= Coutp * Cinp) return;
  int o = g / Cinp, i = g - o * Cinp;
  dst[g] = (o < Cout && i < Cin) ? w[(size_t)o * Cin + i] : 0.0f;
}

__global__ void bn_prep_kernel(const float* __restrict__ gg, const float* __restrict__ bb,
                               const float* __restrict__ mm, const float* __restrict__ vv,
                               int C, float* __restrict__ scale, float* __restrict__ shift) {
  int t = blockIdx.x * blockDim.x + threadIdx.x;
  if (t >= C) return;
  float s = gg[t] / sqrtf(vv[t] + 1e-5f);
  scale[t] = s;
  shift[t] = bb[t] - mm[t] * s;
}

// Brute-force 20-NN on 3-channel canonical points. Insertion sort keeps
// ascending distance; strict '<' reproduces top_k's lower-index tie-break.
__global__ void knn_kernel(const float* __restrict__ pts, int B, int N,
                           int* __restrict__ idx) {
  int t = blockIdx.x * blockDim.x + threadIdx.x;
  if (t >= B * N) return;
  const int BN = B * N;
  int b = t / N;
  const float* px = pts + (size_t)b * N;
  const float* py = px + BN;
  const float* pz = py + BN;
  float qx = pts[t], qy = pts[BN + t], qz = pts[2 * BN + t];
  float bd[KNN]; int bi[KNN];
#pragma unroll
  for (int j = 0; j < KNN; ++j) { bd[j] = 3.4e38f; bi[j] = 0; }
  for (int m = 0; m < N; ++m) {
    float dx = px[m] - qx, dy = py[m] - qy, dz = pz[m] - qz;
    float d = dx * dx + dy * dy + dz * dz;
    if (d < bd[KNN - 1]) {
      bd[KNN - 1] = d; bi[KNN - 1] = m;
#pragma unroll
      for (int j = KNN - 1; j > 0; --j) {
        if (bd[j] < bd[j - 1]) {
          float td = bd[j]; bd[j] = bd[j - 1]; bd[j - 1] = td;
          int ti = bi[j]; bi[j] = bi[j - 1]; bi[j - 1] = ti;
        }
      }
    }
  }
#pragma unroll
  for (int j = 0; j < KNN; ++j) idx[(size_t)t * KNN + j] = bi[j];
}

// Per-point 3x3 covariance over 20 neighbors + closed-form symmetric
// eigenvalues (ascending, matching eigvalsh).
__global__ void cov_eigen_kernel(const float* __restrict__ pts, const int* __restrict__ idx,
                                 int B, int N, float* __restrict__ ev) {
  int t = blockIdx.x * blockDim.x + threadIdx.x;
  if (t >= B * N) return;
  const int BN = B * N;
  int b = t / N;
  const float* px = pts + (size_t)b * N;
  const float* py = px + BN;
  const float* pz = py + BN;
  float sx = 0, sy = 0, sz = 0, sxx = 0, sxy = 0, sxz = 0, syy = 0, syz = 0, szz = 0;
  for (int j = 0; j < KNN; ++j) {
    int n = idx[(size_t)t * KNN + j];
    float x = px[n], y = py[n], z = pz[n];
    sx += x; sy += y; sz += z;
    sxx += x * x; sxy += x * y; sxz += x * z;
    syy += y * y; syz += y * z; szz += z * z;
  }
  const float ik = 1.0f / (float)KNN;
  float mx = sx * ik, my = sy * ik, mz = sz * ik;
  float a00 = sxx * ik - mx * mx, a01 = sxy * ik - mx * my, a02 = sxz * ik - mx * mz;
  float a11 = syy * ik - my * my, a12 = syz * ik - my * mz, a22 = szz * ik - mz * mz;
  float q = (a00 + a11 + a22) * (1.0f / 3.0f);
  float b00 = a00 - q, b11 = a11 - q, b22 = a22 - q;
  float p1 = a01 * a01 + a02 * a02 + a12 * a12;
  float p2 = b00 * b00 + b11 * b11 + b22 * b22 + 2.0f * p1;
  float e0, e1, e2;
  if (p2 <= 1e-30f) {
    e0 = e1 = e2 = q;
  } else {
    float p = sqrtf(p2 * (1.0f / 6.0f));
    float inv = 1.0f / p;
    float c00 = b00 * inv, c01 = a01 * inv, c02 = a02 * inv;
    float c11 = b11 * inv, c12 = a12 * inv, c22 = b22 * inv;
    float detB = c00 * (c11 * c22 - c12 * c12) - c01 * (c01 * c22 - c12 * c02) +
                 c02 * (c01 * c12 - c11 * c02);
    float r = fminf(1.0f, fmaxf(-1.0f, 0.5f * detB));
    float phi = acosf(r) * (1.0f / 3.0f);
    float emax = q + 2.0f * p * cosf(phi);
    float emin = q + 2.0f * p * cosf(phi + 2.0943951023931953f);
    float emid = 3.0f * q - emax - emin;
    e0 = emin; e1 = emid; e2 = emax;
  }
  ev[t] = e0; ev[BN + t] = e1; ev[2 * BN + t] = e2;
}

// Build grouped features: [diff_eu(F), center(F), diff_ei(F), center(F), (dist), pad]
__global__ void build_g_kernel(const float* __restrict__ feat, const float* __restrict__ pts,
                               const int* __restrict__ idx_eu, const int* __restrict__ idx_ei,
                               int F, int B, int N, int Cinp, float* __restrict__ g) {
  const int M = B * N * KNN;
  int col = blockIdx.x * blockDim.x + threadIdx.x;
  if (col >= M) return;
  int t = col / KNN;
  int b = t / N;
  const int BN = B * N;
  int ge = b * N + idx_eu[col];
  int gi = b * N + idx_ei[col];
  for (int f = 0; f < F; ++f) {
    float c = feat[(size_t)f * BN + t];
    float ve = feat[(size_t)f * BN + ge];
    float vi = feat[(size_t)f * BN + gi];
    g[(size_t)f * M + col] = ve - c;
    g[(size_t)(f + F) * M + col] = c;
    g[(size_t)(f + 2 * F) * M + col] = vi - c;
    g[(size_t)(f + 3 * F) * M + col] = c;
  }
  int r = 4 * F;
  if (pts != nullptr) {
    float s = 1e-12f;
#pragma unroll
    for (int c3 = 0; c3 < 3; ++c3) {
      float d = pts[(size_t)c3 * BN + ge] - pts[(size_t)c3 * BN + t];
      s += d * d;
    }
    g[(size_t)r * M + col] = sqrtf(s);
    ++r;
  }
  for (; r < Cinp; ++r) g[(size_t)r * M + col] = 0.0f;
}

// ---------------------------------------------------------------------------
// FP32 WMMA GEMM:  Y[Coutp][M] = act(bn( W[Coutp][Cinp] * X[Cinp][M] ))
// One wave -> 16 rows x 64 cols (4 accumulators reuse the A fragment).
// Requires Cinp % 4 == 0, M % 512 == 0, Coutp % 16 == 0.
// A 16x4 layout: lane<16 -> (row=lane, K=+0/+1); lane>=16 -> (row=lane-16, K=+2/+3)
// B 4x16 layout: v0/v1 hold K rows (kOff,kOff+1), col = lane&15
// C/D: vgpr r -> row = r + (lane>=16 ? 8 : 0), col = lane&15
// ---------------------------------------------------------------------------
__global__ __launch_bounds__(256) void wmma_gemm_kernel(
    const float* __restrict__ Wm, const float* __restrict__ X, float* __restrict__ Y,
    const float* __restrict__ bnscale, const float* __restrict__ bnshift,
    int Cinp, int M, int fuse) {
  const int lane = threadIdx.x & 31;
  const int wave = threadIdx.x >> 5;
  const size_t colBase = ((size_t)blockIdx.x * 8 + wave) * 64;
  const int row0 = blockIdx.y * 16;

  const int aRow = lane & 15;
  const int kOff = (lane >> 4) << 1;  // 0 or 2

  v8f acc0 = {0, 0, 0, 0, 0, 0, 0, 0};
  v8f acc1 = acc0, acc2 = acc0, acc3 = acc0;

  const float* wp = Wm + (size_t)(row0 + aRow) * Cinp + kOff;
  const float* xp = X + colBase + (lane & 15);

  for (int kk = 0; kk < Cinp; kk += 4) {
    v2f a;
    a.x = wp[kk];
    a.y = wp[kk + 1];
    const float* xr0 = xp + (size_t)(kk + kOff) * M;
    const float* xr1 = xr0 + M;
    v2f b0, b1, b2, b3;
    b0.x = xr0[0];  b0.y = xr1[0];
    b1.x = xr0[16]; b1.y = xr1[16];
    b2.x = xr0[32]; b2.y = xr1[32];
    b3.x = xr0[48]; b3.y = xr1[48];
    acc0 = __builtin_amdgcn_wmma_f32_16x16x4_f32(false, a, false, b0, (short)0, acc0, false, false);
    acc1 = __builtin_amdgcn_wmma_f32_16x16x4_f32(false, a, false, b1, (short)0, acc1, false, false);
    acc2 = __builtin_amdgcn_wmma_f32_16x16x4_f32(false, a, false, b2, (short)0, acc2, false, false);
    acc3 = __builtin_amdgcn_wmma_f32_16x16x4_f32(false, a, false, b3, (short)0, acc3, false, false);
  }

  const int rowHalf = (lane >> 4) * 8;
  const size_t col = colBase + (lane & 15);
#pragma unroll
  for (int r = 0; r < 8; ++r) {
    int row = row0 + rowHalf + r;
    float v0 = acc0[r], v1 = acc1[r], v2 = acc2[r], v3 = acc3[r];
    if (fuse) {
      float s = bnscale[row], h = bnshift[row];
      v0 = v0 * s + h; v0 = v0 > 0.0f ? v0 : 0.2f * v0;
      v1 = v1 * s + h; v1 = v1 > 0.0f ? v1 : 0.2f * v1;
      v2 = v2 * s + h; v2 = v2 > 0.0f ? v2 : 0.2f * v2;
      v3 = v3 * s + h; v3 = v3 > 0.0f ? v3 : 0.2f * v3;
    }
    float* yp = Y + (size_t)row * M + col;
    yp[0] = v0; yp[16] = v1; yp[32] = v2; yp[48] = v3;
  }
}

__global__ void maxk_kernel(const float* __restrict__ y, int C, int BN,
                            float* __restrict__ out) {
  size_t g = (size_t)blockIdx.x * blockDim.x + threadIdx.x;
  size_t total = (size_t)C * BN;
  if (g >= total) return;
  int c = (int)(g / BN);
  int t = (int)(g % BN);
  const float* p = y + (size_t)c * BN * KNN + (size_t)t * KNN;
  float m = p[0];
#pragma unroll
  for (int j = 1; j < KNN; ++j) m = fmaxf(m, p[j]);
  out[g] = m;
}

// Farthest point sampling: one workgroup per batch, min-distance array in LDS,
// serial scan with tree argmax (first-index tie-break like jnp.argmax).
__global__ void fps_kernel(const float* __restrict__ pts, int B, int N, int m,
                           int* __restrict__ out) {
  extern __shared__ char smem_raw[];
  float* sd = (float*)smem_raw;
  float* rv = sd + N;
  int* ri = (int*)(rv + blockDim.x);
  int* cur = ri + blockDim.x;
  const int b = blockIdx.x;
  const int tid = threadIdx.x;
  const int BN = B * N;
  const float* px = pts + (size_t)b * N;
  const float* py = px + BN;
  const float* pz = py + BN;
  for (int i = tid; i < N; i += blockDim.x) sd[i] = 1e10f;
  if (tid == 0) { out[(size_t)b * m] = 0; cur[0] = 0; }
  __syncthreads();
  for (int it = 0; it < m - 1; ++it) {
    int last = cur[0];
    float lx = px[last], ly = py[last], lz = pz[last];
    float bestv = -1.0f; int besti = 0;
    for (int i = tid; i < N; i += blockDim.x) {
      float dx = px[i] - lx, dy = py[i] - ly, dz = pz[i] - lz;
      float d = dx * dx + dy * dy + dz * dz;
      float nd = fminf(sd[i], d);
      sd[i] = nd;
      if (nd > bestv) { bestv = nd; besti = i; }
    }
    rv[tid] = bestv; ri[tid] = besti;
    __syncthreads();
    for (int s = blockDim.x >> 1; s > 0; s >>= 1) {
      if (tid < s) {
        float va = rv[tid], vb = rv[tid + s];
        int ia = ri[tid], ib = ri[tid + s];
        if (vb > va || (vb == va && ib < ia)) { rv[tid] = vb; ri[tid] = ib; }
      }
      __syncthreads();
    }
    if (tid == 0) { out[(size_t)b * m + it + 1] = ri[0]; cur[0] = ri[0]; }
    __syncthreads();
  }
}

__global__ void gather_kernel(const float* __restrict__ src, const int* __restrict__ idx,
                              int C, int B, int Ns, int M, float* __restrict__ dst) {
  size_t g = (size_t)blockIdx.x * blockDim.x + threadIdx.x;
  size_t total = (size_t)C * B * M;
  if (g >= total) return;
  size_t BM = (size_t)B * M;
  int c = (int)(g / BM);
  int col = (int)(g % BM);
  int b = col / M, i = col - b * M;
  dst[g] = src[(size_t)c * B * Ns + (size_t)b * Ns + idx[(size_t)b * M + i]];
}

__global__ void three_nn_kernel(const float* __restrict__ tgt, const float* __restrict__ src,
                                int B, int Nt, int Ns, int* __restrict__ idx3,
                                float* __restrict__ w3) {
  int t = blockIdx.x * blockDim.x + threadIdx.x;
  if (t >= B * Nt) return;
  const int BNt = B * Nt, BNs = B * Ns;
  int b = t / Nt;
  float qx = tgt[t], qy = tgt[BNt + t], qz = tgt[2 * BNt + t];
  const float* sx = src + (size_t)b * Ns;
  const float* sy = sx + BNs;
  const float* sz = sy + BNs;
  float d0 = 3.4e38f, d1 = 3.4e38f, d2 = 3.4e38f;
  int i0 = 0, i1 = 0, i2 = 0;
  for (int s = 0; s < Ns; ++s) {
    float dx = sx[s] - qx, dy = sy[s] - qy, dz = sz[s] - qz;
    float d = dx * dx + dy * dy + dz * dz;
    if (d < d2) {
      if (d < d0)      { d2 = d1; i2 = i1; d1 = d0; i1 = i0; d0 = d; i0 = s; }
      else if (d < d1) { d2 = d1; i2 = i1; d1 = d; i1 = s; }
      else             { d2 = d; i2 = s; }
    }
  }
  float r0 = 1.0f / (sqrtf(fmaxf(d0, 0.0f)) + 1e-8f);
  float r1 = 1.0f / (sqrtf(fmaxf(d1, 0.0f)) + 1e-8f);
  float r2 = 1.0f / (sqrtf(fmaxf(d2, 0.0f)) + 1e-8f);
  float inv = 1.0f / (r0 + r1 + r2);
  idx3[(size_t)t * 3 + 0] = i0; w3[(size_t)t * 3 + 0] = r0 * inv;
  idx3[(size_t)t * 3 + 1] = i1; w3[(size_t)t * 3 + 1] = r1 * inv;
  idx3[(size_t)t * 3 + 2] = i2; w3[(size_t)t * 3 + 2] = r2 * inv;
}

__global__ void interp_kernel(const float* __restrict__ feat, const int* __restrict__ idx3,
                              const float* __restrict__ w3, int C, int B, int Ns, int Nt,
                              float* __restrict__ out) {
  size_t g = (size_t)blockIdx.x * blockDim.x + threadIdx.x;
  size_t BNt = (size_t)B * Nt;
  size_t total = (size_t)C * BNt;
  if (g >= total) return;
  int c = (int)(g / BNt);
  int col = (int)(g % BNt);
  int b = col / Nt;
  const float* fb = feat + (size_t)c * B * Ns + (size_t)b * Ns;
  const int* ip = idx3 + (size_t)col * 3;
  const float* wp = w3 + (size_t)col * 3;
  out[g] = wp[0] * fb[ip[0]] + wp[1] * fb[ip[1]] + wp[2] * fb[ip[2]];
}

__global__ void repack_out_kernel(const float* __restrict__ y, float* __restrict__ out,
                                  int B, int N) {
  int g = blockIdx.x * blockDim.x + threadIdx.x;
  int BN = B * N;
  if (g >= 13 * BN) return;
  int o = g / BN, r = g - o * BN;
  int b = r / N, n = r - b * N;
  out[((size_t)b * 13 + o) * N + n] = y[(size_t)o * BN + r];
}

// ---------------------------------------------------------------------------

extern "C" void kernel_launch(void* const* d_in, const int* in_sizes, int n_in,
                              void* d_out, int out_size, void* d_ws, size_t ws_size,
                              hipStream_t stream) {
  (void)in_sizes; (void)n_in; (void)out_size;

  const int B = 4, N1 = 4096, N2 = 2048, N3 = 1024;
  const int BN1 = B * N1, BN2 = B * N2, BN3 = B * N3;
  const int M1 = BN1 * KNN, M2 = BN2 * KNN, M3 = BN3 * KNN;

  const float* x = (const float*)d_in[0];

  char* base = (char*)d_ws;
  size_t off = 0;
  auto alloc = [&](size_t bytes) -> void* {
    off = (off + 255) & ~(size_t)255;
    void* p = base + off;
    off += bytes;
    return p;
  };

  float* p1 = (float*)alloc(sizeof(float) * 3 * BN1);
  float* ev = (float*)alloc(sizeof(float) * 3 * BN1);        // reused per scale
  float* p2 = (float*)alloc(sizeof(float) * 3 * BN2);
  float* p3 = (float*)alloc(sizeof(float) * 3 * BN3);
  int* idx_eu1 = (int*)alloc(sizeof(int) * M1);
  int* idx_ei1 = (int*)alloc(sizeof(int) * M1);
  int* idx_eu2 = (int*)alloc(sizeof(int) * M2);
  int* idx_ei2 = (int*)alloc(sizeof(int) * M2);
  int* idx_eu3 = (int*)alloc(sizeof(int) * M3);
  int* idx_ei3 = (int*)alloc(sizeof(int) * M3);
  int* fps2i = (int*)alloc(sizeof(int) * B * N2);
  int* fps3i = (int*)alloc(sizeof(int) * B * N3);
  float* x1 = (float*)alloc(sizeof(float) * 16 * BN1);
  float* x1d2 = (float*)alloc(sizeof(float) * 16 * BN2);
  float* x2 = (float*)alloc(sizeof(float) * 64 * BN2);
  float* h336 = (float*)alloc(sizeof(float) * 336 * BN3);
  float* h4 = (float*)alloc(sizeof(float) * 1024 * BN3);
  int* nn2i = (int*)alloc(sizeof(int) * BN2 * 3);
  float* nn2w = (float*)alloc(sizeof(float) * BN2 * 3);
  int* nn1i = (int*)alloc(sizeof(int) * BN1 * 3);
  float* nn1w = (float*)alloc(sizeof(float) * BN1 * 3);
  float* w1p = (float*)alloc(sizeof(float) * 16 * 16);
  float* w2p = (float*)alloc(sizeof(float) * 64 * 64);
  float* w3p = (float*)alloc(sizeof(float) * 256 * 256);
  float* w4p = (float*)alloc(sizeof(float) * 1024 * 336);
  float* w5p = (float*)alloc(sizeof(float) * 512 * 1024);
  float* w6p = (float*)alloc(sizeof(float) * 256 * 512);
  float* w7p = (float*)alloc(sizeof(float) * 16 * 256);
  const int couts[6] = {16, 64, 256, 1024, 512, 256};
  float* bns[6]; float* bnh[6];
  for (int l = 0; l < 6; ++l) {
    bns[l] = (float*)alloc(sizeof(float) * couts[l]);
    bnh[l] = (float*)alloc(sizeof(float) * couts[l]);
  }
  const size_t BIGF = 20971520;  // max of 256*M3, 1024*BN1, 64*M2, ...
  float* G  = (float*)alloc(sizeof(float) * BIGF);
  float* YG = (float*)alloc(sizeof(float) * BIGF);
  if (off > ws_size) return;  // workspace too small; nothing safe to do

  auto gridFor = [](size_t n, int bs) { return (unsigned)((n + (size_t)bs - 1) / bs); };
  auto gemm = [&](const float* Wm, int Coutp, int Cinp, const float* X, int M,
                  float* Y, const float* s, const float* h, int fuse) {
    dim3 grid((unsigned)(M / 512), (unsigned)(Coutp / 16));
    wmma_gemm_kernel<<<grid, 256, 0, stream>>>(Wm, X, Y, s, h, Cinp, M, fuse);
  };

  // ---- weights / BN prep -------------------------------------------------
  struct WPack { const float* w; int Cout, Cin, Coutp, Cinp; float* dst; };
  WPack wl[7] = {
      {(const float*)d_in[1], 16, 13, 16, 16, w1p},
      {(const float*)d_in[6], 64, 64, 64, 64, w2p},
      {(const float*)d_in[11], 256, 256, 256, 256, w3p},
      {(const float*)d_in[16], 1024, 336, 1024, 336, w4p},
      {(const float*)d_in[21], 512, 1024, 512, 1024, w5p},
      {(const float*)d_in[26], 256, 512, 256, 512, w6p},
      {(const float*)d_in[31], 13, 256, 16, 256, w7p},
  };
  for (int l = 0; l < 7; ++l) {
    size_t n = (size_t)wl[l].Coutp * wl[l].Cinp;
    pack_weight_kernel<<<gridFor(n, 256), 256, 0, stream>>>(
        wl[l].w, wl[l].Cout, wl[l].Cin, wl[l].Coutp, wl[l].Cinp, wl[l].dst);
  }
  for (int l = 0; l < 6; ++l) {
    int bi = 1 + l * 5;
    bn_prep_kernel<<<gridFor(couts[l], 128), 128, 0, stream>>>(
        (const float*)d_in[bi + 1], (const float*)d_in[bi + 2],
        (const float*)d_in[bi + 3], (const float*)d_in[bi + 4],
        couts[l], bns[l], bnh[l]);
  }

  // ---- scale 1 -----------------------------------------------------------
  repack_pts1_kernel<<<gridFor((size_t)3 * BN1, 256), 256, 0, stream>>>(x, p1, B, N1);
  knn_kernel<<<gridFor(BN1, 256), 256, 0, stream>>>(p1, B, N1, idx_eu1);
  cov_eigen_kernel<<<gridFor(BN1, 256), 256, 0, stream>>>(p1, idx_eu1, B, N1, ev);
  knn_kernel<<<gridFor(BN1, 256), 256, 0, stream>>>(ev, B, N1, idx_ei1);
  build_g_kernel<<<gridFor(M1, 256), 256, 0, stream>>>(ev, p1, idx_eu1, idx_ei1,
                                                       3, B, N1, 16, G);
  gemm(w1p, 16, 16, G, M1, YG, bns[0], bnh[0], 1);
  maxk_kernel<<<gridFor((size_t)16 * BN1, 256), 256, 0, stream>>>(YG, 16, BN1, x1);

  // ---- FPS -> scale 2 ----------------------------------------------------
  size_t smem2 = sizeof(float) * (N1 + 1024) + sizeof(int) * (1024 + 1);
  fps_kernel<<<B, 1024, smem2, stream>>>(p1, B, N1, N2, fps2i);
  gather_kernel<<<gridFor((size_t)3 * BN2, 256), 256, 0, stream>>>(p1, fps2i, 3, B, N1, N2, p2);
  gather_kernel<<<gridFor((size_t)16 * BN2, 256), 256, 0, stream>>>(x1, fps2i, 16, B, N1, N2, x1d2);

  knn_kernel<<<gridFor(BN2, 256), 256, 0, stream>>>(p2, B, N2, idx_eu2);
  cov_eigen_kernel<<<gridFor(BN2, 256), 256, 0, stream>>>(p2, idx_eu2, B, N2, ev);
  knn_kernel<<<gridFor(BN2, 256), 256, 0, stream>>>(ev, B, N2, idx_ei2);
  build_g_kernel<<<gridFor(M2, 256), 256, 0, stream>>>(x1d2, nullptr, idx_eu2, idx_ei2,
                                                       16, B, N2, 64, G);
  gemm(w2p, 64, 64, G, M2, YG, bns[1], bnh[1], 1);
  maxk_kernel<<<gridFor((size_t)64 * BN2, 256), 256, 0, stream>>>(YG, 64, BN2, x2);

  // ---- FPS -> scale 3 ----------------------------------------------------
  size_t smem3 = sizeof(float) * (N2 + 1024) + sizeof(int) * (1024 + 1);
  fps_kernel<<<B, 1024, smem3, stream>>>(p2, B, N2, N3, fps3i);
  gather_kernel<<<gridFor((size_t)3 * BN3, 256), 256, 0, stream>>>(p2, fps3i, 3, B, N2, N3, p3);
  gather_kernel<<<gridFor((size_t)16 * BN3, 256), 256, 0, stream>>>(x1d2, fps3i, 16, B, N2, N3, h336);
  gather_kernel<<<gridFor((size_t)64 * BN3, 256), 256, 0, stream>>>(x2, fps3i, 64, B, N2, N3,
                                                                    h336 + (size_t)16 * BN3);
  float* x2d = h336 + (size_t)16 * BN3;  // [64][BN3]

  knn_kernel<<<gridFor(BN3, 256), 256, 0, stream>>>(p3, B, N3, idx_eu3);
  cov_eigen_kernel<<<gridFor(BN3, 256), 256, 0, stream>>>(p3, idx_eu3, B, N3, ev);
  knn_kernel<<<gridFor(BN3, 256), 256, 0, stream>>>(ev, B, N3, idx_ei3);
  build_g_kernel<<<gridFor(M3, 256), 256, 0, stream>>>(x2d, nullptr, idx_eu3, idx_ei3,
                                                       64, B, N3, 256, G);
  gemm(w3p, 256, 256, G, M3, YG, bns[2], bnh[2], 1);
  maxk_kernel<<<gridFor((size_t)256 * BN3, 256), 256, 0, stream>>>(YG, 256, BN3,
                                                                   h336 + (size_t)80 * BN3);

  // ---- decoder -----------------------------------------------------------
  gemm(w4p, 1024, 336, h336, BN3, h4, bns[3], bnh[3], 1);

  three_nn_kernel<<<gridFor(BN2, 256), 256, 0, stream>>>(p2, p3, B, N2, N3, nn2i, nn2w);
  interp_kernel<<<gridFor((size_t)1024 * BN2, 256), 256, 0, stream>>>(h4, nn2i, nn2w,
                                                                      1024, B, N3, N2, YG);
  three_nn_kernel<<<gridFor(BN1, 256), 256, 0, stream>>>(p1, p2, B, N1, N2, nn1i, nn1w);
  interp_kernel<<<gridFor((size_t)1024 * BN1, 256), 256, 0, stream>>>(YG, nn1i, nn1w,
                                                                      1024, B, N2, N1, G);

  gemm(w5p, 512, 1024, G, BN1, YG, bns[4], bnh[4], 1);   // h5 in YG
  gemm(w6p, 256, 512, YG, BN1, G, bns[5], bnh[5], 1);    // h6 in G
  gemm(w7p, 16, 256, G, BN1, YG, nullptr, nullptr, 0);   // y7 in YG (rows 13..15 are zero)

  repack_out_kernel<<<gridFor((size_t)13 * BN1, 256), 256, 0, stream>>>(YG, (float*)d_out, B, N1);
}